// ExponentialMovingAverage_33904471835537
// MI455X (gfx1250) — compile-verified
//
#include <hip/hip_runtime.h>
#include <hip/hip_bf16.h>
#include <stdint.h>

#define N_TOT   262144
#define K_CODES 1024
#define D_DIM   256
#define DECAYF  0.99f
#define EPSF    1e-5f

#define SLICES  8
#define DSLICE  32          // columns per slice (lane == column == LDS bank group)
#define CHUNKS  16
#define ROWS_PER_CHUNK (N_TOT / CHUNKS)   // 16384
#define TPB     256         // 8 waves of 32
#define DEPTH   8           // outstanding async copies per wave
#define RING    16          // staging ring buffers (slot reuse 8 iters after consume)

typedef __attribute__((ext_vector_type(2))) float v2f;
typedef __attribute__((ext_vector_type(8))) float v8f;

// ---- CDNA5 async global->LDS copy (ASYNCcnt-tracked), ISA 15.18.3 op 96 ----
__device__ __forceinline__ void async_copy_b32(uint32_t lds_off, const float* g) {
  asm volatile("global_load_async_to_lds_b32 %0, %1, off"
               :: "v"(lds_off), "v"((unsigned long long)(uintptr_t)g)
               : "memory");
}
__device__ __forceinline__ void wait_async_le7() {
  asm volatile("s_wait_asynccnt 0x7" ::: "memory");
}

__global__ void __launch_bounds__(TPB) zero_ws_kernel(float* p, int n) {
  int i = blockIdx.x * TPB + threadIdx.x;
  if (i < n) p[i] = 0.0f;
}

// Per-code counts: LDS histogram per chunk, then K global atomics per block.
__global__ void __launch_bounds__(TPB) counts_kernel(const int* __restrict__ indices,
                                                     float* __restrict__ g_cnt) {
  __shared__ float cnt[K_CODES];
  const int tid  = threadIdx.x;
  const int row0 = blockIdx.x * ROWS_PER_CHUNK;
  for (int j = tid; j < K_CODES; j += TPB) cnt[j] = 0.0f;
  __syncthreads();
  for (int r = tid; r < ROWS_PER_CHUNK; r += TPB)
    atomicAdd(&cnt[indices[row0 + r]], 1.0f);     // exact: f32 integers < 2^24
  __syncthreads();
  for (int j = tid; j < K_CODES; j += TPB) atomicAdd(&g_cnt[j], cnt[j]);
}

// Scatter-reduce: encodings stream (async global->LDS, 8-deep) -> LDS acc -> global atomics
__global__ void __launch_bounds__(TPB) scatter_kernel(const int* __restrict__ indices,
                                                      const float* __restrict__ enc,
                                                      float* __restrict__ g_acc) {
  extern __shared__ float lds[];
  float* acc   = lds;                         // K * 32 floats (128 KB)
  float* stage = lds + K_CODES * DSLICE;      // RING * TPB floats (16 KB)

  const int tid   = threadIdx.x;
  const int lane  = tid & 31;
  const int w     = tid >> 5;                 // wave id: one row per wave per iter
  const int chunk = blockIdx.x;
  const int slice = blockIdx.y;

  for (int j = tid; j < K_CODES * DSLICE; j += TPB) acc[j] = 0.0f;
  __syncthreads();

  const int row0  = chunk * ROWS_PER_CHUNK;
  const int ITERS = ROWS_PER_CHUNK / 8;       // 2048 rows per wave
  const float* encBase = enc + (size_t)(row0 + w) * D_DIM + slice * DSLICE + lane;
  const int* idxBase   = indices + row0 + w;  // wave-uniform address stream
  const uint32_t stage_base = (uint32_t)(uintptr_t)(&stage[tid]); // +slot*TPB*4

  // Prime the async pipeline (8 deep)
  for (int p = 0; p < DEPTH; ++p)
    async_copy_b32(stage_base + (uint32_t)(p & (RING - 1)) * (TPB * 4),
                   encBase + (size_t)p * 8 * D_DIM);

  // Steady state: 2 LDS-pipe instructions per iteration (ds_load + ds_add)
  #pragma unroll 4
  for (int i = 0; i < ITERS - DEPTH; ++i) {
    wait_async_le7();                              // oldest of 8 in-flight done
    float v = stage[(i & (RING - 1)) * TPB + tid]; // own slot: no barrier needed
    int idx = idxBase[8 * i];                      // all lanes same addr -> 1 request
    atomicAdd(&acc[idx * DSLICE + lane], v);       // ds_add_f32, bank = lane
    int ip = i + DEPTH;
    async_copy_b32(stage_base + (uint32_t)(ip & (RING - 1)) * (TPB * 4),
                   encBase + (size_t)ip * 8 * D_DIM);
  }
  // Tail: drain the pipeline
  #pragma unroll
  for (int i = ITERS - DEPTH; i < ITERS; ++i) {
    wait_async_le7();
    float v = stage[(i & (RING - 1)) * TPB + tid];
    int idx = idxBase[8 * i];
    atomicAdd(&acc[idx * DSLICE + lane], v);
  }
  __syncthreads();

  // Flush LDS partials to the global accumulator (1 MB, L2-resident)
  for (int j = tid; j < K_CODES * DSLICE; j += TPB) {
    int k = j >> 5, c = j & 31;
    atomicAdd(&g_acc[(size_t)k * D_DIM + slice * DSLICE + c], acc[j]);
  }
}

// new_cs = cs*0.99 + 0.01*counts; n = sum(new_cs) via WMMA (A = ones 16x4).
// Single wave32; EXEC all ones through the WMMA chain.
__global__ void finalize_cs_kernel(const float* __restrict__ cluster_size,
                                   const float* __restrict__ g_cnt,
                                   float* __restrict__ out_cs,
                                   float* __restrict__ n_out) {
  const int lane = threadIdx.x;
  v2f ones; ones.x = 1.0f; ones.y = 1.0f;
  v8f c = {0.f, 0.f, 0.f, 0.f, 0.f, 0.f, 0.f, 0.f};
  for (int t = 0; t < 16; ++t) {
    int b0 = t * 64 + lane, b1 = b0 + 32;
    float n0 = cluster_size[b0] * DECAYF + (1.0f - DECAYF) * g_cnt[b0];
    float n1 = cluster_size[b1] * DECAYF + (1.0f - DECAYF) * g_cnt[b1];
    out_cs[b0] = n0;
    out_cs[b1] = n1;
    v2f bm; bm.x = n0; bm.y = n1;   // any bijective slot layout sums correctly
    // C[i][j] += sum_k B[k][j]; after 16 tiles, row 0 (lanes 0..15 of c[0])
    // holds 16 column-sums whose total is sum(new_cs).
    c = __builtin_amdgcn_wmma_f32_16x16x4_f32(false, ones, false, bm,
                                              (short)0, c, false, false);
  }
  float x = c[0];
  x += __shfl_xor(x, 1, 32);
  x += __shfl_xor(x, 2, 32);
  x += __shfl_xor(x, 4, 32);
  x += __shfl_xor(x, 8, 32);   // lanes 0..15 reduced within their half
  if (lane == 0) n_out[0] = x;
}

// new_ea and embed_normalized; one block per code k, one thread per dim d.
__global__ void __launch_bounds__(D_DIM) finalize_ea_kernel(const float* __restrict__ embed_avg,
                                                            const float* __restrict__ g_acc,
                                                            const float* __restrict__ new_cs,
                                                            const float* __restrict__ n_ptr,
                                                            float* __restrict__ out_ea,
                                                            float* __restrict__ out_norm) {
  int k = blockIdx.x, d = threadIdx.x;
  size_t t = (size_t)k * D_DIM + d;
  float na = embed_avg[t] * DECAYF + (1.0f - DECAYF) * g_acc[t];
  out_ea[t] = na;
  float n  = n_ptr[0];
  float cs = new_cs[k];
  float smoothed = (cs + EPSF) / (n + (float)K_CODES * EPSF) * n;
  out_norm[t] = na / smoothed;
}

extern "C" void kernel_launch(void* const* d_in, const int* in_sizes, int n_in,
                              void* d_out, int out_size, void* d_ws, size_t ws_size,
                              hipStream_t stream) {
  const int*   indices      = (const int*)d_in[0];
  const float* encodings    = (const float*)d_in[1];
  const float* cluster_size = (const float*)d_in[2];
  const float* embed_avg    = (const float*)d_in[3];

  float* out      = (float*)d_out;
  float* out_cs   = out;                             // [K]
  float* out_ea   = out + K_CODES;                   // [K*D]
  float* out_norm = out + K_CODES + K_CODES * D_DIM; // [K*D]

  float* ws    = (float*)d_ws;
  float* g_acc = ws;                          // [K*D] embed_sum accumulator
  float* g_cnt = ws + K_CODES * D_DIM;        // [K]   counts accumulator
  float* n_out = g_cnt + K_CODES;             // [1]   n = sum(new_cs)

  // Zero accumulators every call (required: atomics accumulate across a call).
  int ztot = K_CODES * D_DIM + K_CODES;
  zero_ws_kernel<<<(ztot + TPB - 1) / TPB, TPB, 0, stream>>>(ws, ztot);

  counts_kernel<<<CHUNKS, TPB, 0, stream>>>(indices, g_cnt);

  size_t ldsBytes = (size_t)(K_CODES * DSLICE + RING * TPB) * sizeof(float);
  scatter_kernel<<<dim3(CHUNKS, SLICES), TPB, ldsBytes, stream>>>(indices, encodings, g_acc);

  finalize_cs_kernel<<<1, 32, 0, stream>>>(cluster_size, g_cnt, out_cs, n_out);
  finalize_ea_kernel<<<K_CODES, D_DIM, 0, stream>>>(embed_avg, g_acc, out_cs, n_out,
                                                    out_ea, out_norm);
}